// DotPredictor_54443005444677
// MI455X (gfx1250) — compile-verified
//
#include <hip/hip_runtime.h>

typedef __attribute__((ext_vector_type(2))) float v2f;
typedef __attribute__((ext_vector_type(8))) float v8f;

#define D_FEAT 64

// One wave32 handles 16 edges. For each K-chunk of 4:
//   A (16x4, f32): lane m (0-15) holds h[src[m]][k..k+1] in (v0,v1),
//                  lane m+16     holds h[src[m]][k+2..k+3]      (ISA 7.12.2 layout)
//   B (4x16, f32): lane n (0-15) holds h[dst[n]][k..k+1],
//                  lane n+16     holds h[dst[n]][k+2..k+3]
// => each lane does one float2 gather per matrix per step (min-byte gather).
// Accumulate D (16x16 f32) over 16 steps; diagonal D[n][n] = dot(h[src[n]], h[dst[n]]).
__global__ __launch_bounds__(128) void edge_dot_wmma(
    const float* __restrict__ h,
    const int* __restrict__ src,
    const int* __restrict__ dst,
    float* __restrict__ out,
    int n_edges, int n_groups)
{
    const int lane  = threadIdx.x & 31;
    const int wave  = threadIdx.x >> 5;
    const int group = blockIdx.x * 4 + wave;     // 4 waves per block
    if (group >= n_groups) return;               // wave-uniform guard (before WMMA)

    const int e0     = group << 4;               // first edge of this 16-edge group
    const int r      = lane & 15;                // edge slot this lane serves
    const int colOff = (lane >> 4) << 1;         // 0 for lanes 0-15, 2 for lanes 16-31

    int e = e0 + r;
    if (e >= n_edges) e = n_edges - 1;           // tail clamp (never hit for E=1M)

    const int si = src[e];
    const int di = dst[e];
    const float* __restrict__ srow = h + (size_t)si * D_FEAT + colOff;
    const float* __restrict__ drow = h + (size_t)di * D_FEAT + colOff;

    v8f c = {0.f, 0.f, 0.f, 0.f, 0.f, 0.f, 0.f, 0.f};

#pragma unroll
    for (int k = 0; k < D_FEAT; k += 4) {
        v2f a = *(const v2f*)(srow + k);         // global_load_b64 gather (A matrix)
        v2f b = *(const v2f*)(drow + k);         // global_load_b64 gather (B matrix)
        // 8 args: (neg_a, A, neg_b, B, c_mod, C, reuse_a, reuse_b)
        c = __builtin_amdgcn_wmma_f32_16x16x4_f32(
            /*neg_a=*/false, a, /*neg_b=*/false, b,
            /*c_mod=*/(short)0, c, /*reuse_a=*/false, /*reuse_b=*/false);
    }

    // Diagonal extraction per ISA C/D layout:
    //   VGPR r: lane r      -> D[r][r]
    //           lane r + 24 -> D[r+8][r+8]   (lanes 16-31: M = r+8, N = lane-16)
    float val = 0.f;
    int   n   = -1;
#pragma unroll
    for (int rr = 0; rr < 8; ++rr) {
        if (lane == rr)      { val = c[rr]; n = rr;     }
        if (lane == rr + 24) { val = c[rr]; n = rr + 8; }
    }

    if (n >= 0) {
        const int eo = e0 + n;
        if (eo < n_edges) out[eo] = val;
    }
}

extern "C" void kernel_launch(void* const* d_in, const int* in_sizes, int n_in,
                              void* d_out, int out_size, void* d_ws, size_t ws_size,
                              hipStream_t stream) {
    const float* h   = (const float*)d_in[0];
    const int*   src = (const int*)d_in[1];
    const int*   dst = (const int*)d_in[2];
    float*       out = (float*)d_out;

    const int E       = in_sizes[1];             // number of edges
    const int nGroups = (E + 15) / 16;           // 16 edges per wave
    const int blocks  = (nGroups + 3) / 4;       // 4 waves (128 threads) per block

    edge_dot_wmma<<<blocks, 128, 0, stream>>>(h, src, dst, out, E, nGroups);
}